// ErrorSimulator_30520037605554
// MI455X (gfx1250) — compile-verified
//
#include <hip/hip_runtime.h>

// Shapes fixed by the reference: B=64, H=32, W=32, C=128, NUM_INJ_SITES=256.
#define HWC            (32 * 32 * 128)             // 131072 floats per image
#define F4_PER_IMG     (HWC / 4)                   // 32768 float4 per image
#define THREADS        256                         // 8 wave32 per block
#define F4_PER_THREAD  4
#define F4_PER_BLOCK   (THREADS * F4_PER_THREAD)   // 1024 float4 per block
#define BLOCKS_X       (F4_PER_IMG / F4_PER_BLOCK) // 32 blocks per image

__device__ __forceinline__ float4 fma4(const float4 a, const float4 m, const float4 s) {
    float4 r;
    r.x = fmaf(a.x, m.x, s.x);
    r.y = fmaf(a.y, m.y, s.y);
    r.z = fmaf(a.z, m.z, s.z);
    r.w = fmaf(a.w, m.w, s.w);
    return r;
}

__global__ __launch_bounds__(THREADS) void ErrorSimulator_30520037605554_kernel(
    const float4* __restrict__ inputs,   // [B * F4_PER_IMG]
    const float4* __restrict__ sites,    // [256 * F4_PER_IMG]
    const float4* __restrict__ masks,    // [256 * F4_PER_IMG]
    const int*    __restrict__ idxs,     // [B]
    float4*       __restrict__ out)      // [B * F4_PER_IMG]
{
    const unsigned b     = blockIdx.y;           // batch image (uniform)
    const unsigned chunk = blockIdx.x;           // chunk within image (uniform)
    const int      g     = idxs[b];              // uniform -> s_load_b32 + SGPR base math

    const size_t in_base  = (size_t)b * F4_PER_IMG + (size_t)chunk * F4_PER_BLOCK;
    const size_t gat_base = (size_t)g * F4_PER_IMG + (size_t)chunk * F4_PER_BLOCK;
    const unsigned t = threadIdx.x;

    // Branchless gfx1250 prefetch (global_prefetch_b8) one chunk ahead;
    // last chunk clamps onto itself (harmless, lines already in flight).
    {
        const unsigned nchunk = (chunk + 1u < BLOCKS_X) ? (chunk + 1u) : chunk; // s_cselect
        const size_t in_pf  = (size_t)b * F4_PER_IMG + (size_t)nchunk * F4_PER_BLOCK + t;
        const size_t gat_pf = (size_t)g * F4_PER_IMG + (size_t)nchunk * F4_PER_BLOCK + t;
        __builtin_prefetch(&inputs[in_pf],  0, 1);
        __builtin_prefetch(&masks [gat_pf], 0, 1);
        __builtin_prefetch(&sites [gat_pf], 0, 1);
    }

    // 12 in-flight global_load_b128 (offsets fold into the 24-bit ioffset:
    // i*THREADS float4 = i*4KB), then 16 v_fma_f32 and 4 global_store_b128.
    float4 x[F4_PER_THREAD], m[F4_PER_THREAD], s[F4_PER_THREAD];
#pragma unroll
    for (int i = 0; i < F4_PER_THREAD; ++i) {
        x[i] = inputs[in_base  + i * THREADS + t];
        m[i] = masks [gat_base + i * THREADS + t];
        s[i] = sites [gat_base + i * THREADS + t];
    }
#pragma unroll
    for (int i = 0; i < F4_PER_THREAD; ++i) {
        out[in_base + i * THREADS + t] = fma4(x[i], m[i], s[i]);
    }
}

extern "C" void kernel_launch(void* const* d_in, const int* in_sizes, int n_in,
                              void* d_out, int out_size, void* d_ws, size_t ws_size,
                              hipStream_t stream) {
    (void)n_in; (void)out_size; (void)d_ws; (void)ws_size;

    const float4* inputs = (const float4*)d_in[0];   // [B,H,W,C] f32
    const float4* sites  = (const float4*)d_in[1];   // [256,H,W,C] f32
    const float4* masks  = (const float4*)d_in[2];   // [256,H,W,C] f32
    const int*    idxs   = (const int*)d_in[3];      // [B] i32
    float4*       out    = (float4*)d_out;

    const int B = in_sizes[3];                       // 64 in the reference

    dim3 grid(BLOCKS_X, B, 1);
    dim3 block(THREADS, 1, 1);
    ErrorSimulator_30520037605554_kernel<<<grid, block, 0, stream>>>(
        inputs, sites, masks, idxs, out);
}